// MyNet_31413390803530
// MI455X (gfx1250) — compile-verified
//
#include <hip/hip_runtime.h>
#include <math.h>

typedef float v2f __attribute__((ext_vector_type(2)));
typedef float v8f __attribute__((ext_vector_type(8)));

#define N_NODES     50000
#define NUM_FEATURE 256
#define NUM_HIDDEN  128
#define NUM_CLASSES 16
#define N_EDGES     1600000
#define E_TOT       (N_EDGES + N_NODES)   // edges + self loops

// ---------------------------------------------------------------- utilities

__global__ void zero_u32(unsigned* __restrict__ p, int n) {
    int i = blockIdx.x * blockDim.x + threadIdx.x;
    if (i < n) p[i] = 0u;
}

__global__ void count_deg(const long long* __restrict__ dst,
                          unsigned* __restrict__ cnt, int E) {
    int e = blockIdx.x * blockDim.x + threadIdx.x;
    if (e < E) atomicAdd(&cnt[(int)dst[e]], 1u);
}

__global__ void inv_sqrt_deg(const unsigned* __restrict__ cnt,
                             float* __restrict__ dis, int n) {
    int i = blockIdx.x * blockDim.x + threadIdx.x;
    // +1.0f accounts for the self loop; degree is therefore always > 0.
    if (i < n) dis[i] = rsqrtf((float)cnt[i] + 1.0f);
}

// out[i*C + c] = b[c]   (C is a power of two: 128 or 16)
__global__ void init_bias(float* __restrict__ out, const float* __restrict__ b,
                          int total, int C) {
    int i = blockIdx.x * blockDim.x + threadIdx.x;
    if (i < total) out[i] = b[i & (C - 1)];
}

// ------------------------------------------------------ WMMA fp32 GEMM
// C[M x N] = (RELU_IN ? relu(A) : A)[M x K] @ B[K x N]
// One wave owns a 16-row strip of C and all N/16 column tiles, accumulating
// with V_WMMA_F32_16X16X4_F32 (full fp32, K-step = 4). A fragment is loaded
// once per k-step and reused across all N-tiles.
template <int K, int N, bool RELU_IN>
__global__ void gemm_wmma_f32(const float* __restrict__ A,
                              const float* __restrict__ B,
                              float* __restrict__ C, int M) {
    constexpr int NT = N / 16;
    int wavesPerBlk = (int)(blockDim.x >> 5);
    int strip = (int)blockIdx.x * wavesPerBlk + ((int)threadIdx.x >> 5);
    if (strip >= M / 16) return;

    int lane = (int)threadIdx.x & 31;
    int half = lane >> 4;    // 0: lanes 0-15, 1: lanes 16-31
    int lid  = lane & 15;
    int m0   = strip * 16;

    v8f zero = {};
    v8f acc[NT];
#pragma unroll
    for (int t = 0; t < NT; ++t) acc[t] = zero;

    const float* arow = A + (size_t)(m0 + lid) * K;

    for (int k0 = 0; k0 < K; k0 += 4) {
        // A 16x4 fp32 fragment: lane lid = row M, VGPR pair = K = k0+2*half, +1
        int ka = k0 + half * 2;
        float a0 = arow[ka];
        float a1 = arow[ka + 1];
        if (RELU_IN) { a0 = fmaxf(a0, 0.0f); a1 = fmaxf(a1, 0.0f); }
        v2f av; av.x = a0; av.y = a1;
#pragma unroll
        for (int t = 0; t < NT; ++t) {
            // B 4x16 fragment: lane lid = column N, VGPR v holds K = k0+2*half+v
            v2f bv;
            bv.x = B[(size_t)ka * N + t * 16 + lid];
            bv.y = B[(size_t)(ka + 1) * N + t * 16 + lid];
            acc[t] = __builtin_amdgcn_wmma_f32_16x16x4_f32(
                false, av, false, bv, (short)0, acc[t], false, false);
        }
    }

    // C/D layout: VGPR v -> row m0 + v + 8*half, column n0 + lid
#pragma unroll
    for (int t = 0; t < NT; ++t) {
#pragma unroll
        for (int v = 0; v < 8; ++v) {
            int m = m0 + v + half * 8;
            C[(size_t)m * N + t * 16 + lid] = acc[t][v];
        }
    }
}

// ---------------------------------------------- edge aggregation (layer 1)
// One wave per edge; 128 features -> 4 coalesced float atomics per lane.
__global__ void aggregate128(const float* __restrict__ h,
                             const long long* __restrict__ srcE,
                             const long long* __restrict__ dstE,
                             const float* __restrict__ dis,
                             float* __restrict__ out, int E, int Etot) {
    int gid  = blockIdx.x * blockDim.x + threadIdx.x;
    int edge = gid >> 5;
    int lane = gid & 31;
    if (edge >= Etot) return;
    int s, d;
    if (edge < E) { s = (int)srcE[edge]; d = (int)dstE[edge]; }
    else          { s = d = edge - E; }                    // self loop
    float norm = dis[s] * dis[d];
    const float* hp = h   + (size_t)s * NUM_HIDDEN;
    float*       op = out + (size_t)d * NUM_HIDDEN;
#pragma unroll
    for (int i = 0; i < 4; ++i) {
        int f = lane + i * 32;
        atomicAdd(&op[f], hp[f] * norm);
    }
}

// ---------------------------------------------- edge aggregation (layer 2)
// Thread per (edge, class): 16 classes.
__global__ void aggregate16(const float* __restrict__ z,
                            const long long* __restrict__ srcE,
                            const long long* __restrict__ dstE,
                            const float* __restrict__ dis,
                            float* __restrict__ out, int E, int Etot) {
    int idx = blockIdx.x * blockDim.x + threadIdx.x;
    int e = idx >> 4;
    if (e >= Etot) return;
    int c = idx & 15;
    int s, d;
    if (e < E) { s = (int)srcE[e]; d = (int)dstE[e]; }
    else       { s = d = e - E; }
    float norm = dis[s] * dis[d];
    atomicAdd(&out[(size_t)d * NUM_CLASSES + c],
              z[(size_t)s * NUM_CLASSES + c] * norm);
}

// ------------------------------------------------------------ log-softmax
__global__ void logsoftmax16(float* __restrict__ out, int M) {
    int r = blockIdx.x * blockDim.x + threadIdx.x;
    if (r >= M) return;
    float* p = out + (size_t)r * NUM_CLASSES;
    float v[NUM_CLASSES];
#pragma unroll
    for (int c = 0; c < NUM_CLASSES; ++c) v[c] = p[c];
    float mx = v[0];
#pragma unroll
    for (int c = 1; c < NUM_CLASSES; ++c) mx = fmaxf(mx, v[c]);
    float sum = 0.0f;
#pragma unroll
    for (int c = 0; c < NUM_CLASSES; ++c) sum += expf(v[c] - mx);
    float lse = mx + logf(sum);
#pragma unroll
    for (int c = 0; c < NUM_CLASSES; ++c) p[c] = v[c] - lse;
}

// ---------------------------------------------------------------- launcher

extern "C" void kernel_launch(void* const* d_in, const int* in_sizes, int n_in,
                              void* d_out, int out_size, void* d_ws, size_t ws_size,
                              hipStream_t stream) {
    const float*     x    = (const float*)d_in[0];
    const long long* edge = (const long long*)d_in[1];   // int64, shape (2, E)
    const long long* srcE = edge;                        // edge_index[0]
    const long long* dstE = edge + N_EDGES;              // edge_index[1]
    const float*     W1   = (const float*)d_in[2];
    const float*     b1   = (const float*)d_in[3];
    const float*     W2   = (const float*)d_in[4];
    const float*     b2   = (const float*)d_in[5];
    float*           out  = (float*)d_out;               // [50000, 16]

    // workspace carve-out (~55 MB)
    char* ws = (char*)d_ws;
    unsigned* cnt = (unsigned*)ws;  ws += (size_t)N_NODES * sizeof(unsigned);
    float*    dis = (float*)ws;     ws += (size_t)N_NODES * sizeof(float);
    float*    h   = (float*)ws;     ws += (size_t)N_NODES * NUM_HIDDEN * sizeof(float);
    float*    a1  = (float*)ws;     ws += (size_t)N_NODES * NUM_HIDDEN * sizeof(float);
    float*    z   = (float*)ws;     // [N_NODES, 16]

    // 1) symmetric-normalization coefficients
    zero_u32<<<(N_NODES + 255) / 256, 256, 0, stream>>>(cnt, N_NODES);
    count_deg<<<(N_EDGES + 255) / 256, 256, 0, stream>>>(dstE, cnt, N_EDGES);
    inv_sqrt_deg<<<(N_NODES + 255) / 256, 256, 0, stream>>>(cnt, dis, N_NODES);

    // 2) h = x @ W1   (WMMA fp32, 3125 strips, 4 waves/block)
    int strips = N_NODES / 16;
    gemm_wmma_f32<NUM_FEATURE, NUM_HIDDEN, false>
        <<<(strips + 3) / 4, 128, 0, stream>>>(x, W1, h, N_NODES);

    // 3) a1 = b1; scatter messages (dominant memory phase)
    init_bias<<<(N_NODES * NUM_HIDDEN + 255) / 256, 256, 0, stream>>>(
        a1, b1, N_NODES * NUM_HIDDEN, NUM_HIDDEN);
    aggregate128<<<(E_TOT + 7) / 8, 256, 0, stream>>>(
        h, srcE, dstE, dis, a1, N_EDGES, E_TOT);

    // 4) z = relu(a1) @ W2   (ReLU fused into A-fragment load)
    gemm_wmma_f32<NUM_HIDDEN, NUM_CLASSES, true>
        <<<(strips + 3) / 4, 128, 0, stream>>>(a1, W2, z, N_NODES);

    // 5) out = b2; scatter layer-2 messages directly into d_out
    init_bias<<<(N_NODES * NUM_CLASSES + 255) / 256, 256, 0, stream>>>(
        out, b2, N_NODES * NUM_CLASSES, NUM_CLASSES);
    aggregate16<<<(E_TOT * 16 + 255) / 256, 256, 0, stream>>>(
        z, srcE, dstE, dis, out, N_EDGES, E_TOT);

    // 6) row-wise log-softmax in place
    logsoftmax16<<<(N_NODES + 255) / 256, 256, 0, stream>>>(out, N_NODES);
}